// ModelNew_80908593922613
// MI455X (gfx1250) — compile-verified
//
#include <hip/hip_runtime.h>

typedef float v2f __attribute__((ext_vector_type(2)));
typedef float v8f __attribute__((ext_vector_type(8)));
typedef int   v4i __attribute__((ext_vector_type(4)));

#define AS1 __attribute__((address_space(1)))
#define AS3 __attribute__((address_space(3)))

#define BATCH 16
#define CIN   3
#define DD    32
#define HH    64
#define WW    64
#define OCH   24
#define OD    30
#define OH    62
#define OW    62
#define KTOT  81                    // Cin * 27
#define KPAD  84                    // multiple of 4
#define SPATIAL (OD*OH*OW)          // 115320
#define PLANE (HH*WW)               // 4096
#define SLAB  (CIN*3*PLANE)         // 36864 floats = 144 KB (9 planes)
#define SLABP (SLAB + 64)           // + zeroed pad for edge-column reads

#if defined(__HIP_DEVICE_COMPILE__) && __has_builtin(__builtin_amdgcn_global_load_async_to_lds_b128)
#define HAVE_ASYNC_LDS 1
#else
#define HAVE_ASYNC_LDS 0
#endif

// One block = one (b, od); 8 waves. Wave w: ow-tile (w&3), oh rows (w>>2)::2.
// Stage x[b, :, od:od+3, :, :] (147 KB) in LDS via async-to-LDS, then run an
// implicit-im2col GEMM D[32oc x 16n] += W[32x4] * X[4x16] per K-step with
// V_WMMA_F32_16X16X4_F32, accumulating GroupNorm sum/sumsq in registers.
// Conv activations never touch memory.
__global__ __launch_bounds__(256) void conv_gn_stats(
    const float* __restrict__ x, const float* __restrict__ w,
    const float* __restrict__ bias,
    float* __restrict__ wsum, float* __restrict__ wsq)
{
    __shared__ float Xlds[SLABP];       // [ci*3+kd][64][64] input slab (+pad)
    __shared__ float Wlds[32 * KPAD];   // [oc][k], zero-padded rows/cols
    __shared__ int   tab[KPAD];         // k -> offset into slab (k>=81 -> 0)
    __shared__ float blds[32];          // conv bias, zero-padded

    const int tid = threadIdx.x;
    const int b  = blockIdx.x / OD;
    const int od = blockIdx.x % OD;

    // ---- stage input slab: 9 planes of 4096 floats, 16B vectors ----
    for (int v = tid; v < SLAB / 4; v += 256) {
        const int p = v >> 10;                  // plane: ci*3 + kd
        const int i = (v & 1023) << 2;          // float offset within plane
        const int ci = p / 3, kd = p % 3;
        const float* gsrc = x + ((long)((b * CIN + ci) * DD + od + kd) << 12) + i;
        float* ldst = &Xlds[(p << 12) + i];
#if HAVE_ASYNC_LDS
        __builtin_amdgcn_global_load_async_to_lds_b128(
            (AS1 v4i*)gsrc, (AS3 v4i*)ldst, 0, 0);
#else
        *(float4*)ldst = *(const float4*)gsrc;
#endif
    }
    if (tid < 64) Xlds[SLAB + tid] = 0.f;       // pad: keep edge reads finite

    // ---- stage weights [oc][k] (zero pad oc>=24, k>=81), offsets, bias ----
    for (int i = tid; i < 32 * KPAD; i += 256) {
        int oc = i / KPAD, k = i % KPAD;
        Wlds[i] = (oc < OCH && k < KTOT) ? w[oc * KTOT + k] : 0.f;
    }
    if (tid < KPAD) {
        int k = tid;
        if (k < KTOT) {
            int ci = k / 27, r = k % 27;
            int kd = r / 9, kh = (r % 9) / 3, kw = k % 3;
            tab[k] = ((ci * 3 + kd) << 12) + kh * WW + kw;
        } else {
            tab[k] = 0;     // padded k: any finite slab value; A-row is zero
        }
    }
    if (tid < 32) blds[tid] = (tid < OCH) ? bias[tid] : 0.f;

#if HAVE_ASYNC_LDS && __has_builtin(__builtin_amdgcn_s_wait_asynccnt)
    __builtin_amdgcn_s_wait_asynccnt(0);
#endif
    __syncthreads();

    const int lane   = tid & 31;
    const int wave   = tid >> 5;        // 0..7
    const int laneHi = lane >> 4;       // K-pair / M+8 row select
    const int n      = lane & 15;       // N column
    const int m      = lane & 15;       // A-fragment M row

    const int owTile = wave & 3;
    const int owb    = owTile * 16;
    const int ohStart = wave >> 2;      // 0 or 1; stride 2 over oh
    const int nvalid = (owTile == 3) ? (OW - 48) : 16;
    const bool nval  = (n < nvalid);

    float rb0[8], rb1[8];
    #pragma unroll
    for (int j = 0; j < 8; ++j) {
        rb0[j] = blds[j + 8 * laneHi];
        rb1[j] = blds[16 + j + 8 * laneHi];
    }

    float s0[8], q0[8], s1[8], q1[8];
    #pragma unroll
    for (int j = 0; j < 8; ++j) { s0[j] = q0[j] = s1[j] = q1[j] = 0.f; }

    for (int oh = ohStart; oh < OH; oh += 2) {
        const int baseL = oh * WW + owb + n;    // slab-relative B base
        v8f c0 = {0.f,0.f,0.f,0.f,0.f,0.f,0.f,0.f};
        v8f c1 = {0.f,0.f,0.f,0.f,0.f,0.f,0.f,0.f};

        #pragma unroll
        for (int ks = 0; ks < KPAD / 4; ++ks) {
            const int kb = ks * 4 + 2 * laneHi;     // this lane's K rows
            v2f bf;
            bf.x = Xlds[baseL + tab[kb]];
            bf.y = Xlds[baseL + tab[kb + 1]];
            v2f a0 = *(const v2f*)&Wlds[m * KPAD + kb];          // oc 0..15
            v2f a1 = *(const v2f*)&Wlds[(m + 16) * KPAD + kb];   // oc 16..31
            c0 = __builtin_amdgcn_wmma_f32_16x16x4_f32(
                     false, a0, false, bf, (short)0, c0, false, false);
            c1 = __builtin_amdgcn_wmma_f32_16x16x4_f32(
                     false, a1, false, bf, (short)0, c1, false, false);
        }

        if (nval) {
            #pragma unroll
            for (int j = 0; j < 8; ++j) {
                float y0 = c0[j] + rb0[j];
                s0[j] += y0; q0[j] += y0 * y0;
                if (laneHi == 0) {                  // tile1 rows oc 16..23 only
                    float y1 = c1[j] + rb1[j];
                    s1[j] += y1; q1[j] += y1 * y1;
                }
            }
        }
    }

    // reduce each C-row across its 16 N-lanes, then one atomic per row
    #pragma unroll
    for (int j = 0; j < 8; ++j) {
        float a = s0[j], qq = q0[j];
        #pragma unroll
        for (int msk = 1; msk < 16; msk <<= 1) {
            a  += __shfl_xor(a,  msk, 32);
            qq += __shfl_xor(qq, msk, 32);
        }
        if (n == 0) {                   // lane0 -> oc=j, lane16 -> oc=j+8
            int oc = j + 8 * laneHi;
            atomicAdd(&wsum[b * OCH + oc], a);
            atomicAdd(&wsq [b * OCH + oc], qq);
        }
        float a1 = s1[j], q1v = q1[j];
        #pragma unroll
        for (int msk = 1; msk < 16; msk <<= 1) {
            a1  += __shfl_xor(a1,  msk, 32);
            q1v += __shfl_xor(q1v, msk, 32);
        }
        if (n == 0 && laneHi == 0) {    // oc = 16 + j
            int oc = 16 + j;
            atomicAdd(&wsum[b * OCH + oc], a1);
            atomicAdd(&wsq [b * OCH + oc], q1v);
        }
    }
}

// out[b,c] = gamma_c*(mean_c - mean_g)*rsqrt(var_g+eps) + beta_c
__global__ void gn_finalize(const float* __restrict__ wsum, const float* __restrict__ wsq,
                            const float* __restrict__ gw, const float* __restrict__ gb,
                            float* __restrict__ out)
{
    int t = blockIdx.x * blockDim.x + threadIdx.x;
    if (t >= BATCH * OCH) return;
    int b = t / OCH, c = t % OCH;
    int g = c / (OCH / 8);
    float Ss = 0.f, Sq = 0.f;
    #pragma unroll
    for (int cc = 0; cc < 3; ++cc) {
        Ss += wsum[b * OCH + g * 3 + cc];
        Sq += wsq [b * OCH + g * 3 + cc];
    }
    const float Ng = 3.f * (float)SPATIAL;
    float mg  = Ss / Ng;
    float var = Sq / Ng - mg * mg;
    float mc  = wsum[b * OCH + c] / (float)SPATIAL;
    out[t] = gw[c] * (mc - mg) * rsqrtf(var + 1e-5f) + gb[c];
}

extern "C" void kernel_launch(void* const* d_in, const int* in_sizes, int n_in,
                              void* d_out, int out_size, void* d_ws, size_t ws_size,
                              hipStream_t stream)
{
    const float* x  = (const float*)d_in[0];
    const float* w  = (const float*)d_in[1];
    const float* cb = (const float*)d_in[2];
    const float* gw = (const float*)d_in[3];
    const float* gb = (const float*)d_in[4];
    float* out  = (float*)d_out;
    float* wsum = (float*)d_ws;
    float* wsq  = wsum + BATCH * OCH;

    (void)hipMemsetAsync(d_ws, 0, 2 * BATCH * OCH * sizeof(float), stream);

    conv_gn_stats<<<dim3(BATCH * OD), dim3(256), 0, stream>>>(x, w, cb, wsum, wsq);
    gn_finalize<<<dim3((BATCH * OCH + 255) / 256), dim3(256), 0, stream>>>(
        wsum, wsq, gw, gb, out);
}